// HybridQLSTMTagger_65481071410363
// MI455X (gfx1250) — compile-verified
//
#include <hip/hip_runtime.h>
#include <hip/hip_bf16.h>
#include <math.h>

#define NQ   4
#define DIMQ 16
#define E_DIM 1024
#define IN_W 1032   // E + NQ(probs) + NQ(hx)
#define NT   64

typedef __attribute__((ext_vector_type(2))) float v2f;
typedef __attribute__((ext_vector_type(8))) float v8f;

// Composed CNOT permutation C[k] = perm0[perm1[perm2[perm3[k]]]]
// cnot(c,t): if bit (k>>(3-c))&1 set, flip bit (3-t).
__device__ __forceinline__ int compose_perm(int k) {
    int j = k;
    if (j & 1)        j ^= 8;   // perm3: cnot(3,0)
    if ((j >> 1) & 1) j ^= 1;   // perm2: cnot(2,3)
    if ((j >> 2) & 1) j ^= 2;   // perm1: cnot(1,2)
    if ((j >> 3) & 1) j ^= 4;   // perm0: cnot(0,1)
    return j;
}

__device__ __forceinline__ float2 cmul(float2 a, float2 b) {
    return make_float2(a.x * b.x - a.y * b.y, a.x * b.y + a.y * b.x);
}

// ---------------- Kernel 0: attention_probs (tiny, 1 thread) ----------------
__global__ void k_probs(const float* __restrict__ rot, const float* __restrict__ ent,
                        float* __restrict__ probs) {
    if (threadIdx.x != 0) return;
    float2 u0[4], u1[4];
    for (int q = 0; q < 4; ++q) {
        float a = rot[3 * q] * 0.5f, b = rot[3 * q + 1] * 0.5f, c = rot[3 * q + 2] * 0.5f;
        float ca = cosf(a), sa = sinf(a), cb = cosf(b), sb = sinf(b), cc = cosf(c), sc = sinf(c);
        // v0=(ca,0), v1=(0,-sa); w0 = cb*v0 - sb*v1; w1 = sb*v0 + cb*v1
        float2 w0 = make_float2(cb * ca,  sb * sa);
        float2 w1 = make_float2(sb * ca, -cb * sa);
        // u0 = e^{-ic} w0 ; u1 = e^{+ic} w1
        u0[q] = make_float2(cc * w0.x + sc * w0.y, cc * w0.y - sc * w0.x);
        u1[q] = make_float2(cc * w1.x - sc * w1.y, cc * w1.y + sc * w1.x);
    }
    float2 psi[16];
    for (int k = 0; k < 16; ++k) {
        float2 amp = make_float2(1.f, 0.f);
        for (int q = 0; q < 4; ++q) {
            int bit = (k >> (3 - q)) & 1;
            amp = cmul(amp, bit ? u1[q] : u0[q]);
        }
        psi[k] = amp;
    }
    // entangling controlled-RX on (i, i+1)
    for (int i = 0; i < 3; ++i) {
        float th = ent[i] * 0.5f;
        float ct = cosf(th), st = sinf(th);
        int bi_ = 3 - i, bj = 3 - (i + 1);
        for (int k = 0; k < 16; ++k) {
            if (((k >> bi_) & 1) == 1 && ((k >> bj) & 1) == 0) {
                int k11 = k | (1 << bj);
                float2 p10 = psi[k], p11 = psi[k11];
                // t0 = ct*p10 - i*st*p11 ; t1 = -i*st*p10 + ct*p11
                psi[k]   = make_float2(ct * p10.x + st * p11.y, ct * p10.y - st * p11.x);
                psi[k11] = make_float2(st * p10.y + ct * p11.x, -st * p10.x + ct * p11.y);
            }
        }
    }
    float pr[4] = {0.f, 0.f, 0.f, 0.f};
    for (int k = 0; k < 16; ++k) {
        float pp = psi[k].x * psi[k].x + psi[k].y * psi[k].y;
        for (int w = 0; w < 4; ++w)
            if ((k >> (3 - w)) & 1) pr[w] += pp;
    }
    for (int w = 0; w < 4; ++w) probs[w] = pr[w];
}

// ---------------- Kernel 1: input GEMM via f32 WMMA ----------------
// pre[t][n] = emb[sentence[t]] . Wstk[n][:1024] + probs . Wstk[n][1024:1028] + b[n]
// n = gate*4 + row, gates = {f, i, u, o}. One wave per 16-row tile; N = 16 exactly.
__global__ void k_gemm(const int* __restrict__ sentence, const float* __restrict__ emb,
                       const float* __restrict__ Wf, const float* __restrict__ Wi,
                       const float* __restrict__ Wu, const float* __restrict__ Wo,
                       const float* __restrict__ bf, const float* __restrict__ bi,
                       const float* __restrict__ bu, const float* __restrict__ bo,
                       const float* __restrict__ probs, float* __restrict__ pre) {
    int lane  = threadIdx.x & 31;
    int wv    = threadIdx.x >> 5;
    int tile  = blockIdx.x * 4 + wv;
    int row   = lane & 15;     // M for A, N for B
    int khalf = lane >> 4;     // 0 -> K{0,1}, 1 -> K{2,3}

    int tok = sentence[tile * 16 + row];
    const float* arow = emb + (size_t)tok * E_DIM + khalf * 2;

    int g = row >> 2, r = row & 3;
    const float* wg = (g == 0) ? Wf : (g == 1) ? Wi : (g == 2) ? Wu : Wo;
    const float* bg = (g == 0) ? bf : (g == 1) ? bi : (g == 2) ? bu : bo;
    const float* brow = wg + (size_t)r * IN_W + khalf * 2;

    v8f acc = {0.f, 0.f, 0.f, 0.f, 0.f, 0.f, 0.f, 0.f};
    for (int k = 0; k < E_DIM; k += 4) {
        v2f a = *(const v2f*)(arow + k);
        v2f b = *(const v2f*)(brow + k);
        acc = __builtin_amdgcn_wmma_f32_16x16x4_f32(false, a, false, b,
                                                    (short)0, acc, false, false);
    }

    float bias = bg[r];
    #pragma unroll
    for (int q = 0; q < 4; ++q) bias += probs[q] * wg[(size_t)r * IN_W + E_DIM + q];

    int t0 = tile * 16;
    #pragma unroll
    for (int v = 0; v < 8; ++v) {
        int m = v + khalf * 8;                       // C/D layout: VGPR v -> M=v (+8 for hi lanes)
        pre[(size_t)(t0 + m) * 16 + row] = acc[v] + bias;
    }
}

// ---------------- Kernel 2: serial LSTM scan, one wave ----------------
__global__ void k_scan(const float* __restrict__ pre,
                       const float* __restrict__ Wf, const float* __restrict__ Wi,
                       const float* __restrict__ Wu, const float* __restrict__ Wo,
                       const float* __restrict__ thf, const float* __restrict__ thi,
                       const float* __restrict__ thu, const float* __restrict__ tho,
                       float* __restrict__ hs, int S) {
    int n = threadIdx.x;                 // 0..31 ; lanes 0..15 active logically
    int g = (n >> 2) & 3, r = n & 3;
    const float* wg  = (g == 0) ? Wf  : (g == 1) ? Wi  : (g == 2) ? Wu  : Wo;
    const float* thg = (g == 0) ? thf : (g == 1) ? thi : (g == 2) ? thu : tho;

    float wh[4] = {0.f, 0.f, 0.f, 0.f};
    float th = 0.f;
    if (n < 16) {
        th = thg[r];
        #pragma unroll
        for (int j = 0; j < 4; ++j) wh[j] = wg[(size_t)r * IN_W + E_DIM + NQ + j];
    }

    float hh0 = 0.f, hh1 = 0.f, hh2 = 0.f, hh3 = 0.f;
    float cst = 0.f;                     // lane h (0..3) owns c[h]
    float zc = (n < 16) ? pre[n] : 0.f;

    for (int t = 0; t < S; ++t) {
        float zn = 0.f;                  // prefetch next step's pre row
        if (t + 1 < S && n < 16) zn = pre[(size_t)(t + 1) * 16 + n];

        float z = zc + hh0 * wh[0] + hh1 * wh[1] + hh2 * wh[2] + hh3 * wh[3];
        float ang = 0.5f * (z + th);

        float c2[4], s2[4];
        int base = g * 4;
        #pragma unroll
        for (int q = 0; q < 4; ++q) {
            float a = __shfl(ang, base + q, 32);
            float cs = cosf(a), sn = sinf(a);
            c2[q] = cs * cs; s2[q] = sn * sn;
        }

        float out = 0.f;
        #pragma unroll
        for (int k = 0; k < 16; ++k) {
            int j = compose_perm(k);
            float p = ((j >> 3) & 1 ? s2[0] : c2[0]) * ((j >> 2) & 1 ? s2[1] : c2[1])
                    * ((j >> 1) & 1 ? s2[2] : c2[2]) * ((j     ) & 1 ? s2[3] : c2[3]);
            out += ((k >> (3 - r)) & 1) ? -p : p;
        }

        float act = (g == 2) ? tanhf(out) : 1.f / (1.f + expf(-out));

        float fv = __shfl(act, r,      32);
        float iv = __shfl(act, 4 + r,  32);
        float uv = __shfl(act, 8 + r,  32);
        float ov = __shfl(act, 12 + r, 32);
        float cn = fv * cst + iv * uv;
        float hn = ov * tanhf(cn);
        cst = cn;
        if (n < 4) hs[(size_t)t * 4 + n] = hn;

        hh0 = __shfl(hn, 0, 32);
        hh1 = __shfl(hn, 1, 32);
        hh2 = __shfl(hn, 2, 32);
        hh3 = __shfl(hn, 3, 32);
        zc = zn;
    }
}

// ---------------- Kernel 3: logits + log_softmax, one wave per row ----------------
__global__ void k_logits(const float* __restrict__ hs, const float* __restrict__ W2t,
                         const float* __restrict__ b2t, float* __restrict__ out, int S) {
    int lane = threadIdx.x & 31;
    int wv   = threadIdx.x >> 5;
    int t    = blockIdx.x * 8 + wv;
    if (t >= S) return;
    float h0 = hs[(size_t)t * 4 + 0], h1 = hs[(size_t)t * 4 + 1];
    float h2 = hs[(size_t)t * 4 + 2], h3 = hs[(size_t)t * 4 + 3];
    float l0, l1;
    {
        const float* w = W2t + (size_t)lane * 4;
        l0 = h0 * w[0] + h1 * w[1] + h2 * w[2] + h3 * w[3] + b2t[lane];
    }
    {
        const float* w = W2t + (size_t)(lane + 32) * 4;
        l1 = h0 * w[0] + h1 * w[1] + h2 * w[2] + h3 * w[3] + b2t[lane + 32];
    }
    float m = fmaxf(l0, l1);
    #pragma unroll
    for (int off = 16; off > 0; off >>= 1) m = fmaxf(m, __shfl_xor(m, off, 32));
    float e = expf(l0 - m) + expf(l1 - m);
    #pragma unroll
    for (int off = 16; off > 0; off >>= 1) e += __shfl_xor(e, off, 32);
    float lse = m + logf(e);
    out[(size_t)t * NT + lane]      = l0 - lse;
    out[(size_t)t * NT + lane + 32] = l1 - lse;
}

extern "C" void kernel_launch(void* const* d_in, const int* in_sizes, int n_in,
                              void* d_out, int out_size, void* d_ws, size_t ws_size,
                              hipStream_t stream) {
    (void)n_in; (void)out_size; (void)ws_size;
    const int*   sentence = (const int*)  d_in[0];
    const float* emb = (const float*)d_in[1];
    const float* Wf  = (const float*)d_in[2];
    const float* bf  = (const float*)d_in[3];
    const float* Wi  = (const float*)d_in[4];
    const float* bi  = (const float*)d_in[5];
    const float* Wu  = (const float*)d_in[6];
    const float* bu  = (const float*)d_in[7];
    const float* Wo  = (const float*)d_in[8];
    const float* bo  = (const float*)d_in[9];
    const float* thf = (const float*)d_in[10];
    const float* thi = (const float*)d_in[11];
    const float* thu = (const float*)d_in[12];
    const float* tho = (const float*)d_in[13];
    const float* W2t = (const float*)d_in[14];
    const float* b2t = (const float*)d_in[15];
    const float* rot = (const float*)d_in[16];
    const float* ent = (const float*)d_in[17];

    int S = in_sizes[0];                       // 4096
    char* ws = (char*)d_ws;
    float* probs = (float*)ws;                 // 4 floats (padded to 256 B)
    float* pre   = (float*)(ws + 256);         // S*16 floats
    float* hsbuf = (float*)(ws + 256 + (size_t)S * 16 * sizeof(float)); // S*4 floats

    k_probs<<<1, 32, 0, stream>>>(rot, ent, probs);

    int tiles = S / 16;                        // 256
    k_gemm<<<tiles / 4, 128, 0, stream>>>(sentence, emb, Wf, Wi, Wu, Wo,
                                          bf, bi, bu, bo, probs, pre);

    k_scan<<<1, 32, 0, stream>>>(pre, Wf, Wi, Wu, Wo, thf, thi, thu, tho, hsbuf, S);

    k_logits<<<S / 8, 256, 0, stream>>>(hsbuf, W2t, b2t, (float*)d_out, S);
}